// GNNSolverPolicy_74947179315201
// MI455X (gfx1250) — compile-verified
//
#include <hip/hip_runtime.h>
#include <hip/hip_bf16.h>

// ---- problem constants ----
#define HID    64
#define INDIM  6
// ---- capacity caps (in-degree ~ Binomial(1.2M, 1e-5), mean 12; caps are huge margins) ----
#define MAXR   256      // unique rows needing h2 (agent + its in-neighbors)
#define MAXE1  16384    // edges into the row set (layer-1 aggregation)
#define MAXE2  4096     // edges into the agent (layer-2 aggregation)

typedef __attribute__((ext_vector_type(2))) float v2f;
typedef __attribute__((ext_vector_type(8))) float v8f;

struct Ws {
    float* deg;        // [N]   degree incl. self-loop
    int*   mark;       // [N]   node -> row slot (-1 none, -2 claimed/overflow)
    int*   uniqueR;    // [MAXR]
    int*   E1src;      // [MAXE1]
    int*   E1slot;     // [MAXE1]
    float* E1norm;     // [MAXE1]
    int*   L2src;      // [MAXE2]
    float* L2norm;     // [MAXE2]
    float* out1acc;    // [MAXR*64]  layer-1 pre-relu output rows
    float* h2rows;     // [MAXR*64]  relu(out1) @ W2
    float* out2agent;  // [64]
    int*   ctr;        // [0]=agent [1]=cntR [2]=cntE1 [3]=cntE2
};

__global__ void k_init(Ws w, int n) {
    int i = blockIdx.x * blockDim.x + threadIdx.x;
    if (i < n) { w.deg[i] = 1.0f; w.mark[i] = -1; }
    if (i == 0) { w.ctr[0] = 0x7FFFFFFF; w.ctr[1] = 0; w.ctr[2] = 0; w.ctr[3] = 0; }
}

__global__ void k_deg(Ws w, const int* ei, int E) {
    int e = blockIdx.x * blockDim.x + threadIdx.x;
    if (e < E) atomicAdd(&w.deg[ei[E + e]], 1.0f);
}

__global__ void k_find_agent(Ws w, const float* x, int n) {
    int i = blockIdx.x * blockDim.x + threadIdx.x;
    if (i < n && x[i * INDIM + 1] == 1.0f) atomicMin(&w.ctr[0], i);
}

__global__ void k_seed(Ws w) {
    int a = w.ctr[0];
    w.uniqueR[0] = a;
    w.mark[a] = 0;
    w.ctr[1] = 1;
}

// edges into the agent: record (src, norm) for layer-2 aggregation, and dedupe srcs into uniqueR
__global__ void k_collect_L2(Ws w, const int* ei, int E) {
    int e = blockIdx.x * blockDim.x + threadIdx.x;
    if (e >= E) return;
    int a = w.ctr[0];
    int dst = ei[E + e];
    if (dst != a) return;
    int src = ei[e];
    float nrm = (1.0f / sqrtf(w.deg[src])) * (1.0f / sqrtf(w.deg[dst]));
    int t = atomicAdd(&w.ctr[3], 1);
    if (t < MAXE2) { w.L2src[t] = src; w.L2norm[t] = nrm; }
    int old = atomicCAS(&w.mark[src], -1, -2);
    if (old == -1) {
        int s = atomicAdd(&w.ctr[1], 1);
        if (s < MAXR) { w.uniqueR[s] = src; w.mark[src] = s; }  // finalized at kernel boundary
    }
}

// edges into any node of the row set: needed for layer-1 aggregation
__global__ void k_collect_E1(Ws w, const int* ei, int E) {
    int e = blockIdx.x * blockDim.x + threadIdx.x;
    if (e >= E) return;
    int dst = ei[E + e];
    int m = w.mark[dst];
    if (m < 0 || m >= MAXR) return;
    int src = ei[e];
    float nrm = (1.0f / sqrtf(w.deg[src])) * (1.0f / sqrtf(w.deg[dst]));
    int t = atomicAdd(&w.ctr[2], 1);
    if (t < MAXE1) { w.E1src[t] = src; w.E1slot[t] = m; w.E1norm[t] = nrm; }
}

// out1[slot] = (x[node] @ W1) / deg[node] + b1   (self-loop term + bias)
__global__ void k_out1_self(Ws w, const float* x, const float* W1, const float* b1) {
    int s = blockIdx.x, j = threadIdx.x;
    int cnt = min(w.ctr[1], MAXR);
    if (s >= cnt) return;
    int node = w.uniqueR[s];
    float h = 0.0f;
    #pragma unroll
    for (int k = 0; k < INDIM; ++k) h += x[node * INDIM + k] * W1[k * HID + j];
    w.out1acc[s * HID + j] = h * (1.0f / w.deg[node]) + b1[j];
}

// out1[slot] += (x[src] @ W1) * norm   for each collected edge
__global__ void k_out1_edges(Ws w, const float* x, const float* W1) {
    int t = blockIdx.x, j = threadIdx.x;
    int cnt = min(w.ctr[2], MAXE1);
    if (t >= cnt) return;
    int src = w.E1src[t];
    int slot = w.E1slot[t];
    float nrm = w.E1norm[t];
    float h = 0.0f;
    #pragma unroll
    for (int k = 0; k < INDIM; ++k) h += x[src * INDIM + k] * W1[k * HID + j];
    atomicAdd(&w.out1acc[slot * HID + j], h * nrm);
}

// h2rows = relu(out1acc) @ W2 via V_WMMA_F32_16X16X4_F32.
// One wave per 16-row tile; 4 column tiles x 16 K-steps = 64 WMMA issues.
// A 16x4 f32 layout (ISA 7.12.2): lanes 0-15 hold {K0,K1}, lanes 16-31 hold {K2,K3}, M = lane&15.
// B 4x16 / C,D 16x16 mirrored.
// Padding rows are handled branch-free: the A row index is clamped in-bounds
// (cntR >= 1 always: agent is row 0) and the fragment is scaled by a 0/1 lane
// mask, so EXEC stays all-ones and A loads stay clause-able b64 loads.
__global__ void __launch_bounds__(32) k_h2_wmma(Ws w, const float* W2) {
    int cnt = min(w.ctr[1], MAXR);
    int row0 = blockIdx.x * 16;
    if (row0 >= cnt) return;                 // uniform across the wave
    int lane = threadIdx.x;
    int half = lane >> 4;
    int l16  = lane & 15;
    int rowA = row0 + l16;
    float amask = (rowA < cnt) ? 1.0f : 0.0f;
    int rowC = min(rowA, cnt - 1);           // always a legal row to load
    const v2f* arow = (const v2f*)&w.out1acc[rowC * HID];   // K-pairs are contiguous

    v8f acc[4];
    #pragma unroll
    for (int nt = 0; nt < 4; ++nt)
        #pragma unroll
        for (int r = 0; r < 8; ++r) acc[nt][r] = 0.0f;

    #pragma unroll
    for (int ks = 0; ks < 16; ++ks) {
        int k0 = 4 * ks + 2 * half;          // this lane's K pair within the K-step
        v2f av = arow[k0 >> 1];              // one global_load_b64
        v2f a;
        a.x = fmaxf(av.x, 0.0f) * amask;     // relu + zero-pad, branch-free
        a.y = fmaxf(av.y, 0.0f) * amask;
        #pragma unroll
        for (int nt = 0; nt < 4; ++nt) {
            v2f b;
            b.x = W2[(k0)     * HID + nt * 16 + l16];
            b.y = W2[(k0 + 1) * HID + nt * 16 + l16];
            acc[nt] = __builtin_amdgcn_wmma_f32_16x16x4_f32(
                false, a, false, b, (short)0, acc[nt], false, false);
        }
    }

    // D layout: VGPR r -> M = r (lanes 0-15) / M = 8+r (lanes 16-31), N = nt*16 + l16
    #pragma unroll
    for (int r = 0; r < 8; ++r) {
        int row = row0 + r + 8 * half;
        #pragma unroll
        for (int nt = 0; nt < 4; ++nt)
            w.h2rows[row * HID + nt * 16 + l16] = acc[nt][r];
    }
}

// out2[agent] = h2[agent]/deg[agent] + b2 + sum over agent's in-edges of h2[src]*norm
__global__ void k_out2_agent(Ws w, const float* b2) {
    int j = threadIdx.x;             // 64 threads
    int a = w.ctr[0];
    float v = w.h2rows[0 * HID + j] * (1.0f / w.deg[a]) + b2[j];
    int cnt = min(w.ctr[3], MAXE2);
    for (int t = 0; t < cnt; ++t) {
        int slot = w.mark[w.L2src[t]];
        if (slot >= 0 && slot < MAXR)
            v += w.h2rows[slot * HID + j] * w.L2norm[t];
    }
    w.out2agent[j] = v;
}

__global__ void k_heads(Ws w, const float* Wp, const float* bp,
                        const float* Wv, const float* bv, float* out) {
    int t = threadIdx.x;
    if (t < 4) {
        float s = bp[t];
        for (int j = 0; j < HID; ++j) s += fmaxf(w.out2agent[j], 0.0f) * Wp[j * 4 + t];
        out[t] = s;
    } else if (t == 4) {
        float s = bv[0];
        for (int j = 0; j < HID; ++j) s += fmaxf(w.out2agent[j], 0.0f) * Wv[j];
        out[4] = s;
    }
}

extern "C" void kernel_launch(void* const* d_in, const int* in_sizes, int n_in,
                              void* d_out, int out_size, void* d_ws, size_t ws_size,
                              hipStream_t stream) {
    const float* x  = (const float*)d_in[0];
    const int*   ei = (const int*)  d_in[1];   // [2, E] row-major: src row then dst row
    const float* W1 = (const float*)d_in[2];
    const float* b1 = (const float*)d_in[3];
    const float* W2 = (const float*)d_in[4];
    const float* b2 = (const float*)d_in[5];
    const float* Wp = (const float*)d_in[6];
    const float* bp = (const float*)d_in[7];
    const float* Wv = (const float*)d_in[8];
    const float* bv = (const float*)d_in[9];
    float* out = (float*)d_out;

    const int N = in_sizes[0] / INDIM;
    const int E = in_sizes[1] / 2;

    // carve workspace (256B-aligned slices)
    char* base = (char*)d_ws;
    size_t off = 0;
    auto alloc = [&](size_t bytes) -> void* {
        void* p = base + off;
        off = (off + bytes + 255) & ~(size_t)255;
        return p;
    };
    Ws w;
    w.deg       = (float*)alloc((size_t)N * 4);
    w.mark      = (int*)  alloc((size_t)N * 4);
    w.uniqueR   = (int*)  alloc((size_t)MAXR * 4);
    w.E1src     = (int*)  alloc((size_t)MAXE1 * 4);
    w.E1slot    = (int*)  alloc((size_t)MAXE1 * 4);
    w.E1norm    = (float*)alloc((size_t)MAXE1 * 4);
    w.L2src     = (int*)  alloc((size_t)MAXE2 * 4);
    w.L2norm    = (float*)alloc((size_t)MAXE2 * 4);
    w.out1acc   = (float*)alloc((size_t)MAXR * HID * 4);
    w.h2rows    = (float*)alloc((size_t)MAXR * HID * 4);
    w.out2agent = (float*)alloc(64 * 4);
    w.ctr       = (int*)  alloc(16 * 4);
    if (off > ws_size) return;   // workspace too small: leave output poisoned (visible failure)

    const int TB = 256;
    int gN = (N + TB - 1) / TB;
    int gE = (E + TB - 1) / TB;

    k_init       <<<gN, TB, 0, stream>>>(w, N);
    k_deg        <<<gE, TB, 0, stream>>>(w, ei, E);
    k_find_agent <<<gN, TB, 0, stream>>>(w, x, N);
    k_seed       <<<1, 1, 0, stream>>>(w);
    k_collect_L2 <<<gE, TB, 0, stream>>>(w, ei, E);
    k_collect_E1 <<<gE, TB, 0, stream>>>(w, ei, E);
    k_out1_self  <<<MAXR,  HID, 0, stream>>>(w, x, W1, b1);
    k_out1_edges <<<MAXE1, HID, 0, stream>>>(w, x, W1);
    k_h2_wmma    <<<MAXR / 16, 32, 0, stream>>>(w, W2);
    k_out2_agent <<<1, HID, 0, stream>>>(w, b2);
    k_heads      <<<1, 32, 0, stream>>>(w, Wp, bp, Wv, bv, out);
}